// MaxPool_28656021799353
// MI455X (gfx1250) — compile-verified
//
#include <hip/hip_runtime.h>
#include <cstdint>
#include <cstddef>

// True clang vector type (HIP's float4 is a struct and is rejected by
// __builtin_nontemporal_load/store).
typedef float v4f __attribute__((ext_vector_type(4)));

// Problem shape (fixed by the reference).
constexpr int kB = 4, kH = 128, kW = 128, kC = 256;
constexpr int kHp = kH / 2, kWp = kW / 2;
constexpr long long kOutN = (long long)kB * kHp * kWp * kC;  // 4,194,304 per tensor
constexpr int kThreads = 256;                                 // 8 wave32 waves
constexpr int kPixPerBlk = 4;                                 // 64 lanes (v4f) per pixel
constexpr int kBlocks = (kB * kHp * kWp) / kPixPerBlk;        // 4096

// Order-preserving float <-> uint key (total order, no NaNs in this data path).
__device__ __forceinline__ unsigned orderKey(float f) {
  unsigned u = __float_as_uint(f);
  return (u & 0x80000000u) ? ~u : (u | 0x80000000u);
}
__device__ __forceinline__ float keyToFloat(unsigned u) {
  return __uint_as_float((u & 0x80000000u) ? (u & 0x7FFFFFFFu) : ~u);
}

__global__ void init_reduce_kernel(unsigned* red) {
  red[0] = 0xFF800000u;  // orderKey(+inf): running MIN tracker for u_t
  red[1] = 0x007FFFFFu;  // orderKey(-inf): running MAX tracker for l_t
}

__global__ __launch_bounds__(kThreads) void maxpool_interval_kernel(
    const float* __restrict__ x, const float* __restrict__ bias,
    const float* __restrict__ a, const float* __restrict__ b,
    float* __restrict__ out_x, float* __restrict__ out_bias,
    float* __restrict__ out_a, float* __restrict__ out_b,
    unsigned* __restrict__ red) {
  __shared__ v4f sx[kThreads * 4];      // 16 KB: async-staged x windows
  __shared__ unsigned sred[kThreads];   // 1 KB: block reduction

  const int tid = threadIdx.x;
  const int pix = blockIdx.x * kPixPerBlk + (tid >> 6);  // (b,hp,wp) flat, wp fastest
  const int lane = tid & 63;                             // channel group (4 ch each)
  const int wp = pix & (kWp - 1);
  const int hp = (pix >> 6) & (kHp - 1);
  const int bb = pix >> 12;

  // Element offsets of the four 2x2 window positions for this lane's channels.
  const size_t e00 = ((size_t)((bb * kH + hp * 2) * kW + wp * 2)) * kC + lane * 4;
  const size_t e01 = e00 + kC;
  const size_t e10 = e00 + (size_t)kW * kC;
  const size_t e11 = e10 + kC;

  const unsigned by00 = (unsigned)(e00 * 4u), by01 = (unsigned)(e01 * 4u);
  const unsigned by10 = (unsigned)(e10 * 4u), by11 = (unsigned)(e11 * 4u);

  // CDNA5 async global->LDS staging for x (ASYNCcnt path). Each lane writes its
  // own 4x16B LDS slots; it is also the only reader, so s_wait_asynccnt suffices.
  const unsigned lds0 = (unsigned)(uintptr_t)(&sx[tid * 4]);
  asm volatile("global_load_async_to_lds_b128 %0, %1, %2"
               :: "v"(lds0), "v"(by00), "s"(x) : "memory");
  asm volatile("global_load_async_to_lds_b128 %0, %1, %2"
               :: "v"(lds0 + 16u), "v"(by01), "s"(x) : "memory");
  asm volatile("global_load_async_to_lds_b128 %0, %1, %2"
               :: "v"(lds0 + 32u), "v"(by10), "s"(x) : "memory");
  asm volatile("global_load_async_to_lds_b128 %0, %1, %2"
               :: "v"(lds0 + 48u), "v"(by11), "s"(x) : "memory");

  // Streaming (non-temporal) b128 loads for the gathered tensors; zero reuse,
  // keep them out of the way of the 192MB L2.
  v4f bw[4], aw[4], gw[4];
  bw[0] = __builtin_nontemporal_load((const v4f*)(bias + e00));
  bw[1] = __builtin_nontemporal_load((const v4f*)(bias + e01));
  bw[2] = __builtin_nontemporal_load((const v4f*)(bias + e10));
  bw[3] = __builtin_nontemporal_load((const v4f*)(bias + e11));
  aw[0] = __builtin_nontemporal_load((const v4f*)(a + e00));
  aw[1] = __builtin_nontemporal_load((const v4f*)(a + e01));
  aw[2] = __builtin_nontemporal_load((const v4f*)(a + e10));
  aw[3] = __builtin_nontemporal_load((const v4f*)(a + e11));
  gw[0] = __builtin_nontemporal_load((const v4f*)(b + e00));
  gw[1] = __builtin_nontemporal_load((const v4f*)(b + e01));
  gw[2] = __builtin_nontemporal_load((const v4f*)(b + e10));
  gw[3] = __builtin_nontemporal_load((const v4f*)(b + e11));

  asm volatile("s_wait_asynccnt 0x0" ::: "memory");
  v4f xw[4];
  xw[0] = sx[tid * 4 + 0];
  xw[1] = sx[tid * 4 + 1];
  xw[2] = sx[tid * 4 + 2];
  xw[3] = sx[tid * 4 + 3];

  const float* xf = reinterpret_cast<const float*>(xw);
  const float* bf = reinterpret_cast<const float*>(bw);
  const float* af = reinterpret_cast<const float*>(aw);
  const float* gf = reinterpret_cast<const float*>(gw);

  const float kInf = __uint_as_float(0x7F800000u);
  float umin = kInf, lmax = -kInf;
  v4f rx, rb, ra, rg;

#pragma unroll
  for (int k = 0; k < 4; ++k) {
    // argmax tournament (strict '>' keeps first max: matches jnp.argmax),
    // carrying the three payloads -> pure v_cndmask chains, no scratch.
    float bx = xf[k], sb = bf[k], sa = af[k], sg = gf[k];
#pragma unroll
    for (int j = 1; j < 4; ++j) {
      const float vx = xf[j * 4 + k];
      const bool gt = vx > bx;
      bx = gt ? vx : bx;
      sb = gt ? bf[j * 4 + k] : sb;
      sa = gt ? af[j * 4 + k] : sa;
      sg = gt ? gf[j * 4 + k] : sg;
    }
    rx[k] = bx; rb[k] = sb; ra[k] = sa; rg[k] = sg;

    // Truncated-interval contributions over all 4 window entries.
#pragma unroll
    for (int j = 0; j < 4; ++j) {
      const float tTb = gf[j * 4 + k] - sg;
      const float nom = -((af[j * 4 + k] - sa) + (bf[j * 4 + k] - sb));
      if (tTb > 0.0f)      umin = fminf(umin, nom / tTb);
      else if (tTb < 0.0f) lmax = fmaxf(lmax, nom / tTb);
    }
  }

  const size_t oe = (size_t)pix * kC + lane * 4;
  __builtin_nontemporal_store(rx, (v4f*)(out_x + oe));
  __builtin_nontemporal_store(rb, (v4f*)(out_bias + oe));
  __builtin_nontemporal_store(ra, (v4f*)(out_a + oe));
  __builtin_nontemporal_store(rg, (v4f*)(out_b + oe));

  // Block reduction (order-keys) then one atomic per block per bound.
  sred[tid] = orderKey(umin);
  __syncthreads();
#pragma unroll
  for (int s = kThreads / 2; s > 0; s >>= 1) {
    if (tid < s) sred[tid] = min(sred[tid], sred[tid + s]);
    __syncthreads();
  }
  if (tid == 0) atomicMin(&red[0], sred[0]);
  __syncthreads();

  sred[tid] = orderKey(lmax);
  __syncthreads();
#pragma unroll
  for (int s = kThreads / 2; s > 0; s >>= 1) {
    if (tid < s) sred[tid] = max(sred[tid], sred[tid + s]);
    __syncthreads();
  }
  if (tid == 0) atomicMax(&red[1], sred[0]);
}

__global__ void finalize_kernel(const unsigned* __restrict__ red,
                                const float* __restrict__ l,
                                const float* __restrict__ u,
                                float* __restrict__ out_scalars) {
  out_scalars[0] = fmaxf(l[0], keyToFloat(red[1]));  // l_out
  out_scalars[1] = fminf(u[0], keyToFloat(red[0]));  // u_out
}

extern "C" void kernel_launch(void* const* d_in, const int* in_sizes, int n_in,
                              void* d_out, int out_size, void* d_ws, size_t ws_size,
                              hipStream_t stream) {
  (void)in_sizes; (void)n_in; (void)out_size; (void)ws_size;
  const float* x    = (const float*)d_in[0];
  const float* bias = (const float*)d_in[1];
  const float* a    = (const float*)d_in[2];
  const float* b    = (const float*)d_in[3];
  const float* l    = (const float*)d_in[4];
  const float* u    = (const float*)d_in[5];

  float* out_x    = (float*)d_out;
  float* out_bias = out_x + kOutN;
  float* out_a    = out_x + 2 * kOutN;
  float* out_b    = out_x + 3 * kOutN;
  float* out_lu   = out_x + 4 * kOutN;

  unsigned* red = (unsigned*)d_ws;

  init_reduce_kernel<<<1, 1, 0, stream>>>(red);
  maxpool_interval_kernel<<<kBlocks, kThreads, 0, stream>>>(
      x, bias, a, b, out_x, out_bias, out_a, out_b, red);
  finalize_kernel<<<1, 1, 0, stream>>>(red, l, u, out_lu);
}